// HeteroGINE_71605694759549
// MI455X (gfx1250) — compile-verified
//
#include <hip/hip_runtime.h>

typedef __attribute__((ext_vector_type(2))) float v2f;
typedef __attribute__((ext_vector_type(8))) float v8f;

#define HID 64
#define ED 16
#define LDSTRIDE 68   // 16x64 tile rows padded: bank step 4 -> 16 distinct banks

static __device__ __forceinline__ v8f wmma4(v2f a, v2f b, v8f c) {
  // V_WMMA_F32_16X16X4_F32 : D(16x16,f32) = A(16x4,f32) * B(4x16,f32) + C
  return __builtin_amdgcn_wmma_f32_16x16x4_f32(false, a, false, b, (short)0, c,
                                               false, false);
}

// ---------------------------------------------------------------------------
// out[n_rows, 64] = relu(X[n_rows, Din] @ W[Din, 64] + bias[64])
// one wave -> 16 rows x 64 cols
// ---------------------------------------------------------------------------
__global__ void proj_kernel(const float* __restrict__ X,
                            const float* __restrict__ W,
                            const float* __restrict__ bias,
                            float* __restrict__ out, int n_rows, int Din) {
  const int lane = threadIdx.x & 31;
  const int wave = blockIdx.x * (blockDim.x >> 5) + (threadIdx.x >> 5);
  const long row0 = (long)wave * 16;
  if (row0 >= n_rows) return;                 // wave-uniform exit (EXEC stays full)
  const int half = lane >> 4, l16 = lane & 15;
  long arow = row0 + l16; if (arow >= n_rows) arow = n_rows - 1;
  const float* ap = X + arow * (long)Din + 2 * half;
  v8f acc[4] = {};
  for (int k0 = 0; k0 < Din; k0 += 4) {
    v2f a; a.x = ap[k0]; a.y = ap[k0 + 1];
    const float* bp = W + (long)(k0 + 2 * half) * HID + l16;
#pragma unroll
    for (int t = 0; t < 4; ++t) {
      v2f b; b.x = bp[t * 16]; b.y = bp[t * 16 + HID];
      acc[t] = wmma4(a, b, acc[t]);
    }
  }
#pragma unroll
  for (int t = 0; t < 4; ++t) {
    int col = t * 16 + l16;
    float bv = bias[col];
#pragma unroll
    for (int r = 0; r < 8; ++r) {
      long m = row0 + half * 8 + r;
      if (m < n_rows) {
        float v = acc[t][r] + bv;
        out[m * HID + col] = v > 0.f ? v : 0.f;
      }
    }
  }
}

// ---------------------------------------------------------------------------
// Fused edge stage: msg = relu(h_src[esrc[e]] + ea[e]@We + be)
//                   agg[edst[e]] += msg  (float atomics)
// one wave -> 16 edges; WMMA for ea@We (K=16)
// ---------------------------------------------------------------------------
__global__ void edge_scatter_kernel(const float* __restrict__ ea,
                                    const int* __restrict__ esrc,
                                    const int* __restrict__ edst,
                                    const float* __restrict__ hsrc,
                                    const float* __restrict__ We,
                                    const float* __restrict__ be,
                                    float* __restrict__ agg, int nE) {
  const int lane = threadIdx.x & 31;
  const int wave = blockIdx.x * (blockDim.x >> 5) + (threadIdx.x >> 5);
  const long e0 = (long)wave * 16;
  if (e0 >= nE) return;
  const int half = lane >> 4, l16 = lane & 15;
  long arow = e0 + l16; if (arow >= nE) arow = nE - 1;
  const float* ap = ea + arow * ED + 2 * half;
  v8f acc[4] = {};
#pragma unroll
  for (int k0 = 0; k0 < ED; k0 += 4) {
    v2f a; a.x = ap[k0]; a.y = ap[k0 + 1];
    const float* bp = We + (k0 + 2 * half) * HID + l16;
#pragma unroll
    for (int t = 0; t < 4; ++t) {
      v2f b; b.x = bp[t * 16]; b.y = bp[t * 16 + HID];
      acc[t] = wmma4(a, b, acc[t]);
    }
  }
  int sidx[8], didx[8];
#pragma unroll
  for (int r = 0; r < 8; ++r) {
    long e = e0 + half * 8 + r; if (e >= nE) e = nE - 1;
    sidx[r] = esrc[e];
    didx[r] = edst[e];
  }
#pragma unroll
  for (int t = 0; t < 4; ++t) {
    int col = t * 16 + l16;
    float bv = be[col];
#pragma unroll
    for (int r = 0; r < 8; ++r) {
      long e = e0 + half * 8 + r;
      if (e < nE) {
        float v = acc[t][r] + bv + hsrc[(long)sidx[r] * HID + col];
        v = v > 0.f ? v : 0.f;
        atomicAdd(&agg[(long)didx[r] * HID + col], v);
      }
    }
  }
}

// ---------------------------------------------------------------------------
// Node MLP: h = (1+eps)*hdst + agg ; y = relu(relu(h@W1+b1)@W2+b2)
// out = residual ? hdst + y : y      (safe in-place: per-thread RMW only)
// one wave -> 16 nodes; LDS round-trip converts C-layout -> A-layout
// ---------------------------------------------------------------------------
__global__ void node_update_kernel(const float* __restrict__ hdst,
                                   const float* __restrict__ agg,
                                   const float* __restrict__ W1,
                                   const float* __restrict__ b1,
                                   const float* __restrict__ W2,
                                   const float* __restrict__ b2,
                                   const float* __restrict__ eps_all, int conv,
                                   float* __restrict__ out, int n_rows,
                                   int residual) {
  __shared__ float lds[8 * 16 * LDSTRIDE];
  const int lane = threadIdx.x & 31;
  const int wib = threadIdx.x >> 5;
  const int wave = blockIdx.x * (blockDim.x >> 5) + wib;
  const long row0 = (long)wave * 16;
  if (row0 >= n_rows) return;
  float* buf = lds + wib * 16 * LDSTRIDE;       // wave-private: no barrier needed
  const float epsv = 1.0f + eps_all[conv];

  // stage h tile into LDS in plain [row][col] layout
  for (int i = 0; i < 32; ++i) {
    int idx = lane + 32 * i;
    int rr = idx >> 6, cc = idx & 63;
    long node = row0 + rr;
    float v = 0.f;
    if (node < n_rows) v = epsv * hdst[node * HID + cc] + agg[node * HID + cc];
    buf[rr * LDSTRIDE + cc] = v;
  }

  const int half = lane >> 4, l16 = lane & 15;
  const float* arow = buf + l16 * LDSTRIDE + 2 * half;
  const v8f vzero = {};

  // GEMM 1: h @ W1
  v8f acc[4] = {};
#pragma unroll
  for (int k0 = 0; k0 < HID; k0 += 4) {
    v2f a; a.x = arow[k0]; a.y = arow[k0 + 1];
    const float* bp = W1 + (k0 + 2 * half) * HID + l16;
#pragma unroll
    for (int t = 0; t < 4; ++t) {
      v2f b; b.x = bp[t * 16]; b.y = bp[t * 16 + HID];
      acc[t] = wmma4(a, b, acc[t]);
    }
  }
  // relu(+b1) back to LDS (DS ops are in-order within a wave)
#pragma unroll
  for (int t = 0; t < 4; ++t) {
    int col = t * 16 + l16;
    float bv = b1[col];
#pragma unroll
    for (int r = 0; r < 8; ++r) {
      float v = acc[t][r] + bv;
      buf[(half * 8 + r) * LDSTRIDE + col] = v > 0.f ? v : 0.f;
    }
    acc[t] = vzero;
  }

  // GEMM 2: relu(...) @ W2
#pragma unroll
  for (int k0 = 0; k0 < HID; k0 += 4) {
    v2f a; a.x = arow[k0]; a.y = arow[k0 + 1];
    const float* bp = W2 + (k0 + 2 * half) * HID + l16;
#pragma unroll
    for (int t = 0; t < 4; ++t) {
      v2f b; b.x = bp[t * 16]; b.y = bp[t * 16 + HID];
      acc[t] = wmma4(a, b, acc[t]);
    }
  }
#pragma unroll
  for (int t = 0; t < 4; ++t) {
    int col = t * 16 + l16;
    float bv = b2[col];
#pragma unroll
    for (int r = 0; r < 8; ++r) {
      long m = row0 + half * 8 + r;
      if (m < n_rows) {
        float v = acc[t][r] + bv;
        v = v > 0.f ? v : 0.f;                  // new = relu(gine(...))
        if (residual) v += hdst[m * HID + col]; // h + new
        out[m * HID + col] = v;
      }
    }
  }
}

__global__ void zero4_kernel(float4* __restrict__ p, long n) {
  long i = (long)blockIdx.x * blockDim.x + threadIdx.x;
  if (i < n) p[i] = make_float4(0.f, 0.f, 0.f, 0.f);
}

static inline int blocks_for_rows(long rows) {
  long waves = (rows + 15) / 16;
  return (int)((waves + 7) / 8);   // 8 waves (256 threads) per block
}

extern "C" void kernel_launch(void* const* d_in, const int* in_sizes, int n_in,
                              void* d_out, int out_size, void* d_ws,
                              size_t ws_size, hipStream_t stream) {
  const float* x_user  = (const float*)d_in[0];
  const float* x_item  = (const float*)d_in[1];
  const float* e_attr  = (const float*)d_in[2];
  const int*   src_idx = (const int*)d_in[3];
  const int*   dst_idx = (const int*)d_in[4];
  const float* Wp_u    = (const float*)d_in[5];
  const float* bp_u    = (const float*)d_in[6];
  const float* Wp_i    = (const float*)d_in[7];
  const float* bp_i    = (const float*)d_in[8];
  const float* eps     = (const float*)d_in[9];
  const float* We      = (const float*)d_in[10];
  const float* be      = (const float*)d_in[11];
  const float* W1      = (const float*)d_in[12];
  const float* b1      = (const float*)d_in[13];
  const float* W2      = (const float*)d_in[14];
  const float* b2      = (const float*)d_in[15];

  const int DU = in_sizes[5] / HID;        // 128
  const int DI = in_sizes[7] / HID;        // 64
  const int nU = in_sizes[0] / DU;         // 200000
  const int nI = in_sizes[1] / DI;         // 100000
  const int nE = in_sizes[3];              // 1000000

  float* ws    = (float*)d_ws;
  float* cur_u = ws;                                   // [nU,64] proj output
  float* cur_i = cur_u + (long)nU * HID;               // [nI,64] proj output
  float* agg_u = cur_i + (long)nI * HID;               // [nU,64] aggregate
  float* agg_i = agg_u + (long)nU * HID;               // [nI,64] aggregate
  float* out_u = (float*)d_out;                        // [nU,64] final h_u
  float* out_i = out_u + (long)nU * HID;               // [nI,64] final h_i

  const dim3 B(256);
  const long nzero4 = (long)(nU + nI) * HID / 4;       // agg_u..agg_i contiguous
  const int  zblocks = (int)((nzero4 + 255) / 256);
  const int  gU = blocks_for_rows(nU), gI = blocks_for_rows(nI);
  const int  gE = blocks_for_rows(nE);

  // ---- input projections -------------------------------------------------
  proj_kernel<<<gU, B, 0, stream>>>(x_user, Wp_u, bp_u, cur_u, nU, DU);
  proj_kernel<<<gI, B, 0, stream>>>(x_item, Wp_i, bp_i, cur_i, nI, DI);

  // ---- layer 0 (convs 0 fwd user->item, 1 rev item->user), no residual ---
  zero4_kernel<<<zblocks, B, 0, stream>>>((float4*)agg_u, nzero4);
  edge_scatter_kernel<<<gE, B, 0, stream>>>(e_attr, src_idx, dst_idx, cur_u,
                                            We + 0 * ED * HID, be + 0 * HID,
                                            agg_i, nE);
  edge_scatter_kernel<<<gE, B, 0, stream>>>(e_attr, dst_idx, src_idx, cur_i,
                                            We + 1 * ED * HID, be + 1 * HID,
                                            agg_u, nE);
  node_update_kernel<<<gI, B, 0, stream>>>(cur_i, agg_i,
                                           W1 + 0 * HID * HID, b1 + 0 * HID,
                                           W2 + 0 * HID * HID, b2 + 0 * HID,
                                           eps, 0, out_i, nI, 0);
  node_update_kernel<<<gU, B, 0, stream>>>(cur_u, agg_u,
                                           W1 + 1 * HID * HID, b1 + 1 * HID,
                                           W2 + 1 * HID * HID, b2 + 1 * HID,
                                           eps, 1, out_u, nU, 0);

  // ---- layer 1 (convs 2 fwd, 3 rev), residual, in-place on d_out ---------
  zero4_kernel<<<zblocks, B, 0, stream>>>((float4*)agg_u, nzero4);
  edge_scatter_kernel<<<gE, B, 0, stream>>>(e_attr, src_idx, dst_idx, out_u,
                                            We + 2 * ED * HID, be + 2 * HID,
                                            agg_i, nE);
  edge_scatter_kernel<<<gE, B, 0, stream>>>(e_attr, dst_idx, src_idx, out_i,
                                            We + 3 * ED * HID, be + 3 * HID,
                                            agg_u, nE);
  node_update_kernel<<<gI, B, 0, stream>>>(out_i, agg_i,
                                           W1 + 2 * HID * HID, b1 + 2 * HID,
                                           W2 + 2 * HID * HID, b2 + 2 * HID,
                                           eps, 2, out_i, nI, 1);
  node_update_kernel<<<gU, B, 0, stream>>>(out_u, agg_u,
                                           W1 + 3 * HID * HID, b1 + 3 * HID,
                                           W2 + 3 * HID * HID, b2 + 3 * HID,
                                           eps, 3, out_u, nU, 1);
}